// PriorityEncoder8_23407571764165
// MI455X (gfx1250) — compile-verified
//
#include <hip/hip_runtime.h>
#include <hip/hip_bf16.h>

// 8-bit priority encoder over 4,194,304 rows of 8 float32 {0,1} values.
// out[i] = 1.0f iff bit i is the highest set bit of the row (P7 = MSB priority),
// all zeros if the row is all zero. Pure streaming problem:
//   256 MiB total traffic / 23.3 TB/s  ->  ~11.5 us floor on MI455X.
// Strategy: 128-bit non-temporal loads/stores (read-once / write-once stream,
// don't thrash the 192 MB L2), branchless MSB-isolation in integer VALU,
// global_prefetch_b8 for the next grid-stride tile. WMMA intentionally unused:
// arithmetic intensity is ~0.4 int-ops/byte, so matrix ops cannot help.

typedef unsigned int u32x4 __attribute__((ext_vector_type(4)));

#define ONE_F32_BITS 0x3F800000u

__global__ __launch_bounds__(256, 8)
void prio8_encode_kernel(const u32x4* __restrict__ in,
                         u32x4* __restrict__ out,
                         unsigned nrows) {
    const unsigned stride = blockDim.x * gridDim.x;
    for (unsigned row = blockIdx.x * blockDim.x + threadIdx.x;
         row < nrows; row += stride) {

        // Prefetch next iteration's input line (lowers to global_prefetch_b8).
        unsigned nxt = row + stride;
        if (nxt < nrows) {
            __builtin_prefetch((const void*)&in[2 * (size_t)nxt]);
        }

        // One row = 32 B = two 128-bit words. Non-temporal: read-once stream.
        u32x4 lo = __builtin_nontemporal_load(&in[2 * (size_t)row]);       // P0..P3
        u32x4 hi = __builtin_nontemporal_load(&in[2 * (size_t)row + 1]);   // P4..P7

        // float 1.0f has nonzero bit pattern, 0.0f is all-zero -> word != 0 test.
        unsigned m = (lo.x ? 1u   : 0u) | (lo.y ? 2u   : 0u)
                   | (lo.z ? 4u   : 0u) | (lo.w ? 8u   : 0u)
                   | (hi.x ? 16u  : 0u) | (hi.y ? 32u  : 0u)
                   | (hi.z ? 64u  : 0u) | (hi.w ? 128u : 0u);

        // Isolate highest set bit (one-hot); hb == 0 when m == 0 (all-zero row).
        unsigned s = m | (m >> 1);
        s |= s >> 2;
        s |= s >> 4;
        unsigned hb = s ^ (s >> 1);

        u32x4 olo, ohi;
        olo.x = (hb & 1u)   ? ONE_F32_BITS : 0u;
        olo.y = (hb & 2u)   ? ONE_F32_BITS : 0u;
        olo.z = (hb & 4u)   ? ONE_F32_BITS : 0u;
        olo.w = (hb & 8u)   ? ONE_F32_BITS : 0u;
        ohi.x = (hb & 16u)  ? ONE_F32_BITS : 0u;
        ohi.y = (hb & 32u)  ? ONE_F32_BITS : 0u;
        ohi.z = (hb & 64u)  ? ONE_F32_BITS : 0u;
        ohi.w = (hb & 128u) ? ONE_F32_BITS : 0u;

        // Write-once stream: non-temporal 128-bit stores.
        __builtin_nontemporal_store(olo, &out[2 * (size_t)row]);
        __builtin_nontemporal_store(ohi, &out[2 * (size_t)row + 1]);
    }
}

extern "C" void kernel_launch(void* const* d_in, const int* in_sizes, int n_in,
                              void* d_out, int out_size, void* d_ws, size_t ws_size,
                              hipStream_t stream) {
    (void)n_in; (void)out_size; (void)d_ws; (void)ws_size;

    const u32x4* in  = (const u32x4*)d_in[0];   // P: [nrows, 8] float32 -> 2 x u32x4 per row
    u32x4*       out = (u32x4*)d_out;

    const unsigned nrows = (unsigned)(in_sizes[0] / 8);

    const int threads        = 256;  // 8 wave32 per block
    const int rows_per_thread = 4;   // grid-stride: 4 iterations/thread
    const unsigned per_block = (unsigned)(threads * rows_per_thread);
    const int blocks = (int)((nrows + per_block - 1) / per_block);

    prio8_encode_kernel<<<blocks, threads, 0, stream>>>(in, out, nrows);
}